// GraphNet_74895639707914
// MI455X (gfx1250) — compile-verified
//
#include <hip/hip_runtime.h>
#include <math.h>

// ---------------------------------------------------------------- constants
#define NN   8192      // nodes
#define EE   65536     // original edges
#define ETOT 73728     // edges + self loops
#define BG   128       // graphs
#define GFDIM 5
#define HH   4
#define CC   128

typedef __attribute__((ext_vector_type(2))) float v2f;
typedef __attribute__((ext_vector_type(4))) float v4f;
typedef __attribute__((ext_vector_type(8))) float v8f;

static __device__ __forceinline__ float elu_f(float v)  { return v > 0.f ? v : expm1f(v); }
static __device__ __forceinline__ float lrelu(float v, float s) { return v > 0.f ? v : s * v; }
static __device__ __forceinline__ unsigned ford(float f) {
  unsigned u = __float_as_uint(f);
  return (u & 0x80000000u) ? ~u : (u | 0x80000000u);
}
static __device__ __forceinline__ float funord(unsigned u) {
  return __uint_as_float((u & 0x80000000u) ? (u ^ 0x80000000u) : ~u);
}

// two K=4 WMMAs consuming one b128 fragment pair (K-permuted consistently on A & B)
static __device__ __forceinline__ void mm8(const v4f a, const v4f b, v8f& c) {
  v2f al = {a.x, a.y}, ah = {a.z, a.w};
  v2f bl = {b.x, b.y}, bh = {b.z, b.w};
  c = __builtin_amdgcn_wmma_f32_16x16x4_f32(false, al, false, bl, (short)0, c, false, false);
  c = __builtin_amdgcn_wmma_f32_16x16x4_f32(false, ah, false, bh, (short)0, c, false, false);
}

// ---------------------------------------------------------------- utility
__global__ void k_zero(float* p, long n) {
  long i = (long)blockIdx.x * blockDim.x + threadIdx.x;
  if (i < n) p[i] = 0.f;
}

// self-loop attr = segment_sum(edge_attr over dst)
__global__ void k_selfloop(const float* ea, const int* dst, float* out, int E) {
  int e = blockIdx.x * blockDim.x + threadIdx.x;
  if (e < E) atomicAdd(&out[dst[e]], ea[e]);
}

__global__ void k_build_ea(const float* eattr, const float* loop, float* ea, int E, int ET) {
  int e = blockIdx.x * blockDim.x + threadIdx.x;
  if (e < ET) ea[e] = (e < E) ? eattr[e] : loop[e - E];
}

// t[m,j] = elu(in[m]*w[j]+b[j])   (1-column input MLP)
__global__ void k_premlp(const float* in, const float* w, const float* b, float* t, int M, int J) {
  int i = blockIdx.x * blockDim.x + threadIdx.x;
  if (i < M * J) { int m = i / J, j = i - m * J; t[i] = elu_f(in[m] * w[j] + b[j]); }
}

// BatchNorm (training-mode batch stats, biased var)
__global__ void k_bn_stats(const float* t, float* s, float* sq, int M, int J) {
  int i = blockIdx.x * blockDim.x + threadIdx.x;
  if (i < M * J) { int j = i % J; float v = t[i]; atomicAdd(&s[j], v); atomicAdd(&sq[j], v * v); }
}
__global__ void k_bn_apply(const float* t, const float* s, const float* sq, const float* g,
                           const float* be, float* out, int M, int J, int ldo, int coff) {
  int i = blockIdx.x * blockDim.x + threadIdx.x;
  if (i < M * J) {
    int m = i / J, j = i - m * J;
    float mu = s[j] / (float)M;
    float var = sq[j] / (float)M - mu * mu;
    out[(long)m * ldo + coff + j] = g[j] * (t[i] - mu) * rsqrtf(var + 1e-5f) + be[j];
  }
}

// yh = elu(y @ pg_w.T + pg_b), written straight into the concat buffer
__global__ void k_yh(const float* y, const float* w, const float* b, float* out,
                     int B, int O, int ldo, int coff) {
  int i = blockIdx.x * blockDim.x + threadIdx.x;
  if (i < B * O) {
    int m = i / O, o = i - m * O;
    float a = b[o];
    for (int j = 0; j < GFDIM; ++j) a += y[m * GFDIM + j] * w[o * GFDIM + j];
    out[(long)m * ldo + coff + o] = elu_f(a);
  }
}

// ------------------------------------------------- WMMA f32 GEMM: C = A*W^T + bias
// A: MxK row-major. W: NcolsxK row-major. Requirements: K % 8 == 0 (or K == 8),
// Ncols % 64 == 0, M % 16 == 0. One wave computes a 16x64 tile; per 8-wide K step
// five b128 loads feed eight v_wmma_f32_16x16x4_f32, software-pipelined so next
// iteration's fragments load while current WMMAs execute.
// act: 0 none, 1 elu, 2 leaky(0.01)
__global__ void k_gemm_wmma(const float* __restrict__ A, const float* __restrict__ W,
                            const float* __restrict__ bias, float* __restrict__ C,
                            int M, int K, int Ncols, int act) {
  int lane   = threadIdx.x & 31;
  int wtile  = blockIdx.x * (blockDim.x >> 5) + (threadIdx.x >> 5);
  int tilesN = Ncols >> 6;                     // 64-wide column groups
  int tr     = wtile / tilesN;
  int row0   = tr << 4;
  int col0   = (wtile - tr * tilesN) << 6;
  if (row0 >= M) return;                       // wave-uniform: EXEC stays all-ones

  int ml = lane & 15;                          // A row / B col within tile
  int kh = lane >> 4;                          // low/high K quad select
  const float* arow = A + (size_t)(row0 + ml) * K + 4 * kh;
  const float* w0   = W + (size_t)(col0      + ml) * K + 4 * kh;
  const float* w1   = W + (size_t)(col0 + 16 + ml) * K + 4 * kh;
  const float* w2   = W + (size_t)(col0 + 32 + ml) * K + 4 * kh;
  const float* w3   = W + (size_t)(col0 + 48 + ml) * K + 4 * kh;

  v8f c0 = {0.f,0.f,0.f,0.f,0.f,0.f,0.f,0.f};
  v8f c1 = c0, c2 = c0, c3 = c0;

  // prologue loads (K >= 8 always)
  v4f a  = *(const v4f*)(arow);
  v4f b0 = *(const v4f*)(w0);
  v4f b1 = *(const v4f*)(w1);
  v4f b2 = *(const v4f*)(w2);
  v4f b3 = *(const v4f*)(w3);

  for (int k0 = 8; k0 < K; k0 += 8) {
    // prefetch next fragments before consuming current ones
    v4f an  = *(const v4f*)(arow + k0);
    v4f n0  = *(const v4f*)(w0 + k0);
    v4f n1  = *(const v4f*)(w1 + k0);
    v4f n2  = *(const v4f*)(w2 + k0);
    v4f n3  = *(const v4f*)(w3 + k0);
    mm8(a, b0, c0);
    mm8(a, b1, c1);
    mm8(a, b2, c2);
    mm8(a, b3, c3);
    a = an; b0 = n0; b1 = n1; b2 = n2; b3 = n3;
  }
  mm8(a, b0, c0);
  mm8(a, b1, c1);
  mm8(a, b2, c2);
  mm8(a, b3, c3);

  v8f accs[4] = {c0, c1, c2, c3};
  float* cbase = C + (size_t)(row0 + kh * 8) * Ncols + col0 + ml;
#pragma unroll
  for (int j = 0; j < 4; ++j) {
    float bb = bias ? bias[col0 + j * 16 + ml] : 0.f;
#pragma unroll
    for (int r = 0; r < 8; ++r) {              // vgpr r -> row r + 8*kh, col j*16 + ml
      float v = accs[j][r] + bb;
      if (act == 1)      v = elu_f(v);
      else if (act == 2) v = lrelu(v, 0.01f);
      cbase[(size_t)r * Ncols + j * 16] = v;
    }
  }
}

// ------------------------------------------------- GATv2 edge logits
// logits[e,h] = sum_c leaky0.2(xl[src,h,c]+xr[dst,h,c] (+ a_prev@we)) * att[h,c]
__global__ void k_gat_logits(const float* xl, const float* xr, const float* att,
                             const float* aprev, const float* we,
                             const int* ei, float* logits, int ET, int E) {
  int i = blockIdx.x * blockDim.x + threadIdx.x;
  if (i >= ET * HH) return;
  int e = i >> 2, h = i & 3;
  int s, d;
  if (e < E) { s = ei[e]; d = ei[E + e]; } else { s = d = e - E; }
  const float* pl = xl + (size_t)s * (HH * CC) + h * CC;
  const float* pr = xr + (size_t)d * (HH * CC) + h * CC;
  float a0 = 0, a1 = 0, a2 = 0, a3 = 0;
  if (we) { a0 = aprev[e*4+0]; a1 = aprev[e*4+1]; a2 = aprev[e*4+2]; a3 = aprev[e*4+3]; }
  float acc = 0.f;
  for (int c = 0; c < CC; ++c) {
    float z = pl[c] + pr[c];
    if (we) {
      const float* wr = we + (size_t)(h * CC + c) * 4;
      z += a0 * wr[0] + a1 * wr[1] + a2 * wr[2] + a3 * wr[3];
    }
    acc += lrelu(z, 0.2f) * att[h * CC + c];
  }
  logits[i] = acc;
}

// ------------------------------------------------- segment softmax over dst
__global__ void k_segmax(const float* logits, const int* ei, unsigned* umax, int ET, int E) {
  int i = blockIdx.x * blockDim.x + threadIdx.x;
  if (i >= ET * HH) return;
  int e = i >> 2, h = i & 3;
  int d = (e < E) ? ei[E + e] : (e - E);
  atomicMax(&umax[d * HH + h], ford(logits[i]));
}
__global__ void k_segexp(const float* logits, const int* ei, const unsigned* umax,
                         float* alpha, float* denom, int ET, int E) {
  int i = blockIdx.x * blockDim.x + threadIdx.x;
  if (i >= ET * HH) return;
  int e = i >> 2, h = i & 3;
  int d = (e < E) ? ei[E + e] : (e - E);
  float p = expf(logits[i] - funord(umax[d * HH + h]));
  alpha[i] = p;
  atomicAdd(&denom[d * HH + h], p);
}
__global__ void k_segdiv(float* alpha, const int* ei, const float* denom, int ET, int E) {
  int i = blockIdx.x * blockDim.x + threadIdx.x;
  if (i >= ET * HH) return;
  int e = i >> 2, h = i & 3;
  int d = (e < E) ? ei[E + e] : (e - E);
  alpha[i] /= denom[d * HH + h];
}

// ------------------------------------------------- GAT aggregation (sum over e,h)
__global__ void k_gat_agg(const float* alpha, const float* xl, const int* ei,
                          float* acc, int ET, int E) {
  int i = blockIdx.x * blockDim.x + threadIdx.x;
  if (i >= ET * CC) return;
  int e = i >> 7, c = i & 127;
  int s, d;
  if (e < E) { s = ei[e]; d = ei[E + e]; } else { s = d = e - E; }
  float v = 0.f;
#pragma unroll
  for (int h = 0; h < HH; ++h)
    v += alpha[e * 4 + h] * xl[(size_t)s * (HH * CC) + h * CC + c];
  atomicAdd(&acc[(size_t)d * CC + c], v);
}
__global__ void k_gat_final(const float* acc, const float* bias, float* out, int N) {
  int i = blockIdx.x * blockDim.x + threadIdx.x;
  if (i < N * CC) out[i] = acc[i] * 0.25f + bias[i & 127];
}

// ------------------------------------------------- stable segment product
__global__ void k_segprod_edges(const float* h, const int* ei, float* lg, float* ng,
                                int ET, int E) {
  int i = blockIdx.x * blockDim.x + threadIdx.x;
  if (i >= ET * CC) return;
  int e = i >> 7, c = i & 127;
  int s, d;
  if (e < E) { s = ei[e]; d = ei[E + e]; } else { s = d = e - E; }
  float v = h[(size_t)s * CC + c];
  atomicAdd(&lg[(size_t)d * CC + c], logf(fabsf(v)));
  if (v < 0.f) atomicAdd(&ng[(size_t)d * CC + c], 1.f);
}
__global__ void k_segprod_nodes(const float* h, const int* seg, float* lg, float* ng, int N) {
  int i = blockIdx.x * blockDim.x + threadIdx.x;
  if (i >= N * CC) return;
  int n = i >> 7, c = i & 127;
  int b = seg[n];
  float v = h[i];
  atomicAdd(&lg[b * CC + c], logf(fabsf(v)));
  if (v < 0.f) atomicAdd(&ng[b * CC + c], 1.f);
}
__global__ void k_segprod_final(const float* lg, const float* ng, float* out, long n, int do_leaky) {
  long i = (long)blockIdx.x * blockDim.x + threadIdx.x;
  if (i < n) {
    float sgn = 1.f - 2.f * fmodf(ng[i], 2.f);
    float v = sgn * expf(lg[i]);
    out[i] = do_leaky ? lrelu(v, 0.01f) : v;
  }
}

// ------------------------------------------------- GraphNorm
__global__ void k_gn_sum(const float* h, const int* batch, float* gsum, float* gcnt, int N) {
  int i = blockIdx.x * blockDim.x + threadIdx.x;
  if (i >= N * CC) return;
  int n = i >> 7, c = i & 127;
  atomicAdd(&gsum[batch[n] * CC + c], h[i]);
  if (c == 0) atomicAdd(&gcnt[batch[n]], 1.f);
}
__global__ void k_gn_var(const float* h, const int* batch, const float* gsum, const float* gcnt,
                         const float* a, float* gvar, int N) {
  int i = blockIdx.x * blockDim.x + threadIdx.x;
  if (i >= N * CC) return;
  int n = i >> 7, c = i & 127;
  int b = batch[n];
  float mu = gsum[b * CC + c] / gcnt[b];
  float o = h[i] - a[c] * mu;
  atomicAdd(&gvar[b * CC + c], o * o);
}
__global__ void k_gn_apply(const float* h, const int* batch, const float* gsum, const float* gcnt,
                           const float* a, const float* gvar, const float* w, const float* bb,
                           float* out, int N) {
  int i = blockIdx.x * blockDim.x + threadIdx.x;
  if (i >= N * CC) return;
  int n = i >> 7, c = i & 127;
  int b = batch[n];
  float cnt = gcnt[b];
  float mu = gsum[b * CC + c] / cnt;
  float o = h[i] - a[c] * mu;
  out[i] = w[c] * o * rsqrtf(gvar[b * CC + c] / cnt + 1e-5f) + bb[c];
}

// ------------------------------------------------- edge-embedding input (zero-padded K=8)
__global__ void k_etin(const float* eh, const float* alpha2, float* etin, int ET) {
  int e = blockIdx.x * blockDim.x + threadIdx.x;
  if (e >= ET) return;
  float* o = etin + (size_t)e * 8;
  o[0] = eh[e * 2];         o[1] = eh[e * 2 + 1];
  o[2] = alpha2[e * 4];     o[3] = alpha2[e * 4 + 1];
  o[4] = alpha2[e * 4 + 2]; o[5] = alpha2[e * 4 + 3];
  o[6] = 0.f;               o[7] = 0.f;
}
__global__ void k_poole(const float* et2, const int* eb, float* pool, int ET) {
  int i = blockIdx.x * blockDim.x + threadIdx.x;
  if (i >= ET * CC) return;
  int e = i >> 7, c = i & 127;
  atomicAdd(&pool[eb[e] * CC + c], et2[i]);
}

// ------------------------------------------------- final dot + sigmoid
__global__ void k_f4(const float* o3, const float* w, const float* b, float* out, int B) {
  int bi = blockIdx.x * blockDim.x + threadIdx.x;
  if (bi >= B) return;
  float a = b[0];
  for (int j = 0; j < 256; ++j) a += o3[bi * 256 + j] * w[j];
  out[bi] = 1.f / (1.f + expf(-a));
}

// ================================================================ host launcher
static inline int gdiv(long n, int b) { return (int)((n + b - 1) / b); }

extern "C" void kernel_launch(void* const* d_in, const int* in_sizes, int n_in,
                              void* d_out, int out_size, void* d_ws, size_t ws_size,
                              hipStream_t stream) {
  (void)in_sizes; (void)n_in; (void)out_size; (void)ws_size;
  const float* x       = (const float*)d_in[0];
  const float* eattr   = (const float*)d_in[1];
  const float* y       = (const float*)d_in[2];
  const float* pn_w    = (const float*)d_in[3];
  const float* pn_b    = (const float*)d_in[4];
  const float* pn_g    = (const float*)d_in[5];
  const float* pn_be   = (const float*)d_in[6];
  const float* pe_w    = (const float*)d_in[7];
  const float* pe_b    = (const float*)d_in[8];
  const float* pe_g    = (const float*)d_in[9];
  const float* pe_be   = (const float*)d_in[10];
  const float* pg_w    = (const float*)d_in[11];
  const float* pg_b    = (const float*)d_in[12];
  const float* g1_wl   = (const float*)d_in[13];
  const float* g1_bl   = (const float*)d_in[14];
  const float* g1_wr   = (const float*)d_in[15];
  const float* g1_br   = (const float*)d_in[16];
  const float* g1_att  = (const float*)d_in[17];
  const float* g1_bias = (const float*)d_in[18];
  const float* g2_wl   = (const float*)d_in[19];
  const float* g2_bl   = (const float*)d_in[20];
  const float* g2_wr   = (const float*)d_in[21];
  const float* g2_br   = (const float*)d_in[22];
  const float* g2_we   = (const float*)d_in[23];
  const float* g2_att  = (const float*)d_in[24];
  const float* g2_bias = (const float*)d_in[25];
  const float* gn_w    = (const float*)d_in[26];
  const float* gn_b    = (const float*)d_in[27];
  const float* gn_a    = (const float*)d_in[28];
  const float* ee_w1   = (const float*)d_in[29];
  const float* ee_b1   = (const float*)d_in[30];
  const float* ee_w2   = (const float*)d_in[31];
  const float* ee_b2   = (const float*)d_in[32];
  const float* bnn_g   = (const float*)d_in[33];
  const float* bnn_b   = (const float*)d_in[34];
  const float* bne_g   = (const float*)d_in[35];
  const float* bne_b   = (const float*)d_in[36];
  const float* f_w1    = (const float*)d_in[37];
  const float* f_b1    = (const float*)d_in[38];
  const float* f_w2    = (const float*)d_in[39];
  const float* f_b2    = (const float*)d_in[40];
  const float* f_w3    = (const float*)d_in[41];
  const float* f_b3    = (const float*)d_in[42];
  const float* f_w4    = (const float*)d_in[43];
  const float* f_b4    = (const float*)d_in[44];
  const int*   ei      = (const int*)d_in[45];   // [2,E]
  const int*   batch   = (const int*)d_in[46];
  const int*   ebatch  = (const int*)d_in[47];

  float* ws = (float*)d_ws;
  size_t off = 0;
  auto alloc = [&](size_t n) { float* p = ws + off; off += n; return p; };

  float* loop  = alloc(NN);
  float* ea    = alloc(ETOT);
  float* t_n   = alloc((size_t)NN * 16);
  float* xh    = alloc((size_t)NN * 16);
  float* t_e   = alloc((size_t)ETOT * 2);
  float* eh    = alloc((size_t)ETOT * 2);
  float* stats = alloc(4096);                    // sum | sumsq (<=2048 ch)
  float* xl    = alloc((size_t)NN * 512);        // reused layer1 -> layer2
  float* xr    = alloc((size_t)NN * 512);
  float* lg1   = alloc((size_t)ETOT * 4);        // logits1
  float* al1   = alloc((size_t)ETOT * 4);        // alpha1
  float* lg2   = alloc((size_t)ETOT * 4);
  float* al2   = alloc((size_t)ETOT * 4);
  float* umaxf = alloc((size_t)NN * 4);          // used as unsigned
  float* denom = alloc((size_t)NN * 4);
  float* acc   = alloc((size_t)NN * CC);
  float* hbuf  = alloc((size_t)NN * CC);         // h1 then h2
  float* slg   = alloc((size_t)NN * CC);
  float* sng   = alloc((size_t)NN * CC);
  float* hp    = alloc((size_t)NN * CC);         // h1p then h2p
  float* hn    = alloc((size_t)NN * CC);         // graphnorm out
  float* gsum  = alloc((size_t)BG * CC);
  float* gcnt  = alloc(BG);
  float* gvar  = alloc((size_t)BG * CC);
  float* etin  = alloc((size_t)ETOT * 8);        // 6 used + 2 zero pad (K%8==0)
  float* et1   = alloc((size_t)ETOT * CC);
  float* et2   = alloc((size_t)ETOT * CC);
  float* plg   = alloc((size_t)BG * CC);
  float* png   = alloc((size_t)BG * CC);
  float* pnp   = alloc((size_t)BG * CC);         // node pool (pre-BN)
  float* pep   = alloc((size_t)BG * CC);         // edge pool (pre-BN)
  float* cat   = alloc((size_t)BG * 512);
  float* o1    = alloc((size_t)BG * 2048);
  float* o2    = alloc((size_t)BG * 2048);
  float* o3    = alloc((size_t)BG * 256);

  const int T = 256;
  auto Z = [&](float* p, long n) { k_zero<<<gdiv(n, T), T, 0, stream>>>(p, n); };
  auto GEMM = [&](const float* A, const float* W, const float* b, float* C,
                  int M, int K, int Nc, int act) {
    long tiles = (long)(M / 16) * (Nc / 64);     // 16x64 tile per wave
    k_gemm_wmma<<<gdiv(tiles * 32, 128), 128, 0, stream>>>(A, W, b, C, M, K, Nc, act);
  };

  // ---- preprocessing
  Z(loop, NN);
  k_selfloop<<<gdiv(EE, T), T, 0, stream>>>(eattr, ei + EE, loop, EE);
  k_premlp<<<gdiv((long)NN * 16, T), T, 0, stream>>>(x, pn_w, pn_b, t_n, NN, 16);
  Z(stats, 4096);
  k_bn_stats<<<gdiv((long)NN * 16, T), T, 0, stream>>>(t_n, stats, stats + 2048, NN, 16);
  k_bn_apply<<<gdiv((long)NN * 16, T), T, 0, stream>>>(t_n, stats, stats + 2048, pn_g, pn_be, xh, NN, 16, 16, 0);
  k_build_ea<<<gdiv(ETOT, T), T, 0, stream>>>(eattr, loop, ea, EE, ETOT);
  k_premlp<<<gdiv((long)ETOT * 2, T), T, 0, stream>>>(ea, pe_w, pe_b, t_e, ETOT, 2);
  Z(stats, 4096);
  k_bn_stats<<<gdiv((long)ETOT * 2, T), T, 0, stream>>>(t_e, stats, stats + 2048, ETOT, 2);
  k_bn_apply<<<gdiv((long)ETOT * 2, T), T, 0, stream>>>(t_e, stats, stats + 2048, pe_g, pe_be, eh, ETOT, 2, 2, 0);
  k_yh<<<gdiv((long)BG * 256, T), T, 0, stream>>>(y, pg_w, pg_b, cat, BG, 256, 512, 256);

  // ---- GATv2 layer 1
  GEMM(xh, g1_wl, g1_bl, xl, NN, 16, 512, 0);
  GEMM(xh, g1_wr, g1_br, xr, NN, 16, 512, 0);
  k_gat_logits<<<gdiv((long)ETOT * 4, T), T, 0, stream>>>(xl, xr, g1_att, nullptr, nullptr, ei, lg1, ETOT, EE);
  Z(umaxf, (long)NN * 4); Z(denom, (long)NN * 4);
  k_segmax<<<gdiv((long)ETOT * 4, T), T, 0, stream>>>(lg1, ei, (unsigned*)umaxf, ETOT, EE);
  k_segexp<<<gdiv((long)ETOT * 4, T), T, 0, stream>>>(lg1, ei, (unsigned*)umaxf, al1, denom, ETOT, EE);
  k_segdiv<<<gdiv((long)ETOT * 4, T), T, 0, stream>>>(al1, ei, denom, ETOT, EE);
  Z(acc, (long)NN * CC);
  k_gat_agg<<<gdiv((long)ETOT * CC, T), T, 0, stream>>>(al1, xl, ei, acc, ETOT, EE);
  k_gat_final<<<gdiv((long)NN * CC, T), T, 0, stream>>>(acc, g1_bias, hbuf, NN);
  Z(slg, (long)NN * CC); Z(sng, (long)NN * CC);
  k_segprod_edges<<<gdiv((long)ETOT * CC, T), T, 0, stream>>>(hbuf, ei, slg, sng, ETOT, EE);
  k_segprod_final<<<gdiv((long)NN * CC, T), T, 0, stream>>>(slg, sng, hp, (long)NN * CC, 1);

  // ---- GATv2 layer 2 (edge features = alpha1 @ g2_we.T folded into logits)
  GEMM(hp, g2_wl, g2_bl, xl, NN, CC, 512, 0);
  GEMM(hp, g2_wr, g2_br, xr, NN, CC, 512, 0);
  k_gat_logits<<<gdiv((long)ETOT * 4, T), T, 0, stream>>>(xl, xr, g2_att, al1, g2_we, ei, lg2, ETOT, EE);
  Z(umaxf, (long)NN * 4); Z(denom, (long)NN * 4);
  k_segmax<<<gdiv((long)ETOT * 4, T), T, 0, stream>>>(lg2, ei, (unsigned*)umaxf, ETOT, EE);
  k_segexp<<<gdiv((long)ETOT * 4, T), T, 0, stream>>>(lg2, ei, (unsigned*)umaxf, al2, denom, ETOT, EE);
  k_segdiv<<<gdiv((long)ETOT * 4, T), T, 0, stream>>>(al2, ei, denom, ETOT, EE);
  Z(acc, (long)NN * CC);
  k_gat_agg<<<gdiv((long)ETOT * CC, T), T, 0, stream>>>(al2, xl, ei, acc, ETOT, EE);
  k_gat_final<<<gdiv((long)NN * CC, T), T, 0, stream>>>(acc, g2_bias, hbuf, NN);
  Z(slg, (long)NN * CC); Z(sng, (long)NN * CC);
  k_segprod_edges<<<gdiv((long)ETOT * CC, T), T, 0, stream>>>(hbuf, ei, slg, sng, ETOT, EE);
  k_segprod_final<<<gdiv((long)NN * CC, T), T, 0, stream>>>(slg, sng, hp, (long)NN * CC, 1);

  // ---- GraphNorm
  Z(gsum, (long)BG * CC); Z(gcnt, BG); Z(gvar, (long)BG * CC);
  k_gn_sum<<<gdiv((long)NN * CC, T), T, 0, stream>>>(hp, batch, gsum, gcnt, NN);
  k_gn_var<<<gdiv((long)NN * CC, T), T, 0, stream>>>(hp, batch, gsum, gcnt, gn_a, gvar, NN);
  k_gn_apply<<<gdiv((long)NN * CC, T), T, 0, stream>>>(hp, batch, gsum, gcnt, gn_a, gvar, gn_w, gn_b, hn, NN);

  // ---- edge attention embedding
  k_etin<<<gdiv(ETOT, T), T, 0, stream>>>(eh, al2, etin, ETOT);
  GEMM(etin, ee_w1, ee_b1, et1, ETOT, 8, CC, 1);      // K padded 6 -> 8, elu
  GEMM(et1, ee_w2, ee_b2, et2, ETOT, CC, CC, 0);

  // ---- pooling + BN -> concat
  Z(pep, (long)BG * CC);
  k_poole<<<gdiv((long)ETOT * CC, T), T, 0, stream>>>(et2, ebatch, pep, ETOT);
  Z(stats, 4096);
  k_bn_stats<<<gdiv((long)BG * CC, T), T, 0, stream>>>(pep, stats, stats + 2048, BG, CC);
  k_bn_apply<<<gdiv((long)BG * CC, T), T, 0, stream>>>(pep, stats, stats + 2048, bne_g, bne_b, cat, BG, CC, 512, CC);
  Z(plg, (long)BG * CC); Z(png, (long)BG * CC);
  k_segprod_nodes<<<gdiv((long)NN * CC, T), T, 0, stream>>>(hn, batch, plg, png, NN);
  k_segprod_final<<<gdiv((long)BG * CC, T), T, 0, stream>>>(plg, png, pnp, (long)BG * CC, 0);
  Z(stats, 4096);
  k_bn_stats<<<gdiv((long)BG * CC, T), T, 0, stream>>>(pnp, stats, stats + 2048, BG, CC);
  k_bn_apply<<<gdiv((long)BG * CC, T), T, 0, stream>>>(pnp, stats, stats + 2048, bnn_g, bnn_b, cat, BG, CC, 512, 0);

  // ---- final MLP
  GEMM(cat, f_w1, f_b1, o1, BG, 512, 2048, 1);
  GEMM(o1,  f_w2, f_b2, o2, BG, 2048, 2048, 1);
  GEMM(o2,  f_w3, f_b3, o3, BG, 2048, 256, 1);
  k_f4<<<1, 128, 0, stream>>>(o3, f_w4, f_b4, (float*)d_out, BG);
}